// P2F_dist_9723805958689
// MI455X (gfx1250) — compile-verified
//
#include <hip/hip_runtime.h>
#include <hip/hip_bf16.h>

typedef float v2f __attribute__((ext_vector_type(2)));
typedef float v8f __attribute__((ext_vector_type(8)));

#define P_N 2048
#define F_N 8192
#define CHUNKS 16
#define FACES_PER_CHUNK (F_N / CHUNKS)        // 512
#define TILES_PER_CHUNK (FACES_PER_CHUNK / 16) // 32
#define PT_TILES (P_N / 16)                    // 128
#define WAVES_TOTAL (PT_TILES * CHUNKS)        // 2048

// v_wmma_f32_16x16x4_f32: A = 16x4 f32 (2 VGPR), B = 4x16 f32 (2 VGPR), C/D = 16x16 f32 (8 VGPR)
__device__ __forceinline__ v8f wmma4(v2f a, v2f b, v8f c) {
  return __builtin_amdgcn_wmma_f32_16x16x4_f32(
      /*neg_a=*/false, a, /*neg_b=*/false, b,
      /*c_mod=*/(short)0, c, /*reuse_a=*/false, /*reuse_b=*/false);
}

// fast safe-reciprocal matching reference sdiv (d==0 -> divide by 1)
__device__ __forceinline__ float rcp_s(float d) {
  float ds = (d == 0.0f) ? 1.0f : d;
  return __builtin_amdgcn_rcpf(ds);
}

// ---------------------------------------------------------------------------
// Kernel 1: per-face precompute. 16 floats/face:
// [0..2]=ab  [3..5]=ac  [6..8]=a  [9]=-0.5|a|^2  [10]=-a.ab  [11]=-a.ac
// [12]=|ab|^2  [13]=|ac|^2  [14]=ab.ac  [15]=pad
// ---------------------------------------------------------------------------
__global__ void prep_faces_kernel(const float* __restrict__ verts,
                                  const int* __restrict__ faces,
                                  float* __restrict__ fd) {
  int f = blockIdx.x * blockDim.x + threadIdx.x;
  if (f >= F_N) return;
  int ia = faces[3 * f + 0], ib = faces[3 * f + 1], ic = faces[3 * f + 2];
  float ax = verts[3 * ia], ay = verts[3 * ia + 1], az = verts[3 * ia + 2];
  float bx = verts[3 * ib], by = verts[3 * ib + 1], bz = verts[3 * ib + 2];
  float cx = verts[3 * ic], cy = verts[3 * ic + 1], cz = verts[3 * ic + 2];
  float abx = bx - ax, aby = by - ay, abz = bz - az;
  float acx = cx - ax, acy = cy - ay, acz = cz - az;
  float a_ab = ax * abx + ay * aby + az * abz;
  float a_ac = ax * acx + ay * acy + az * acz;
  float a2   = ax * ax + ay * ay + az * az;
  float Lab  = abx * abx + aby * aby + abz * abz;
  float Lac  = acx * acx + acy * acy + acz * acz;
  float Mm   = abx * acx + aby * acy + abz * acz;
  float* o = fd + (size_t)f * 16;
  float4* o4 = (float4*)o;
  o4[0] = make_float4(abx, aby, abz, acx);
  o4[1] = make_float4(acy, acz, ax, ay);
  o4[2] = make_float4(az, -0.5f * a2, -a_ab, -a_ac);
  o4[3] = make_float4(Lab, Lac, Mm, 0.0f);
}

// ---------------------------------------------------------------------------
// Kernel 2: main sweep. One wave = one (16-point tile, 512-face chunk).
// 3 WMMAs per 16x16 tile produce q = p.a - 0.5|a|^2, d1 = p.ab - a.ab,
// d2 = p.ac - a.ac (constants folded into WMMA C). Then Ericson region
// classification + running min per lane, shfl_xor butterfly over columns.
// ---------------------------------------------------------------------------
__global__ __launch_bounds__(256) void p2f_main_kernel(
    const float* __restrict__ pts,
    const float* __restrict__ fd,
    float* __restrict__ part) {
  int wave = (blockIdx.x * blockDim.x + threadIdx.x) >> 5;
  int lane = threadIdx.x & 31;
  int pt_tile = wave / CHUNKS;
  int chunk = wave % CHUNKS;
  int col = lane & 15;
  bool hi = lane >= 16;

  // A matrix (16x4): row M = col; lanes 0-15 hold {K0=x, K1=y}, 16-31 {K2=z, K3=0}
  int m = pt_tile * 16 + col;
  float px = pts[3 * m], py = pts[3 * m + 1], pz = pts[3 * m + 2];
  v2f A;
  A.x = hi ? pz : px;
  A.y = hi ? 0.0f : py;

  float bs[8], bw2[8], bw3[8];
  int bi[8];
#pragma unroll
  for (int r = 0; r < 8; ++r) { bs[r] = 3.402823466e38f; bi[r] = 0; bw2[r] = 0.f; bw3[r] = 0.f; }

  int fbase = chunk * FACES_PER_CHUNK;
  for (int t = 0; t < TILES_PER_CHUNK; ++t) {
    int fcol = fbase + t * 16 + col;                 // this lane's column face
    const float4* rptr = (const float4*)(fd + (size_t)fcol * 16);
    float4 r0 = rptr[0], r1 = rptr[1], r2 = rptr[2], r3 = rptr[3];

    // B matrices (4x16): VGPR0 lanes0-15 K0 / lanes16-31 K2; VGPR1 K1 / K3(=0)
    v2f Bab, Bac, Ba;
    Bab.x = hi ? r0.z : r0.x;  Bab.y = hi ? 0.0f : r0.y;   // ab
    Bac.x = hi ? r1.y : r0.w;  Bac.y = hi ? 0.0f : r1.x;   // ac
    Ba.x  = hi ? r2.x : r1.z;  Ba.y  = hi ? 0.0f : r1.w;   // a

    // C matrices: per-column constant broadcast over all 8 row VGPRs
    float cq = r2.y, c1 = r2.z, c2 = r2.w;
    v8f Cq = {cq, cq, cq, cq, cq, cq, cq, cq};
    v8f C1 = {c1, c1, c1, c1, c1, c1, c1, c1};
    v8f C2 = {c2, c2, c2, c2, c2, c2, c2, c2};

    v8f Dq = wmma4(A, Ba,  Cq);   // p.a - 0.5|a|^2
    v8f D1 = wmma4(A, Bab, C1);   // d1 = (p-a).ab
    v8f D2 = wmma4(A, Bac, C2);   // d2 = (p-a).ac

    float Lab = r3.x, Lac = r3.y, Mm = r3.z;
#pragma unroll
    for (int r = 0; r < 8; ++r) {
      float q = Dq[r], d1 = D1[r], d2 = D2[r];
      float d3 = d1 - Lab, d4 = d2 - Mm, d5 = d1 - Mm, d6 = d2 - Lac;
      float vc = d1 * d4 - d3 * d2;
      float vb = d5 * d2 - d1 * d6;
      float va = d3 * d6 - d5 * d4;
      float rd = rcp_s(va + vb + vc);
      float w2 = vb * rd, w3 = vc * rd;              // interior (region 0)
      float n6 = d4 - d3, p6 = d5 - d6;
      float t6 = n6 * rcp_s(n6 + p6);
      bool m6 = (va <= 0.f) && (n6 >= 0.f) && (p6 >= 0.f);
      w2 = m6 ? 1.f - t6 : w2;  w3 = m6 ? t6 : w3;   // edge BC
      float t5 = d2 * rcp_s(d2 - d6);
      bool m5 = (vb <= 0.f) && (d2 >= 0.f) && (d6 <= 0.f);
      w2 = m5 ? 0.f : w2;       w3 = m5 ? t5 : w3;   // edge AC
      bool m4 = (d6 >= 0.f) && (d5 <= d6);
      w2 = m4 ? 0.f : w2;       w3 = m4 ? 1.f : w3;  // vertex C
      float t3 = d1 * rcp_s(d1 - d3);
      bool m3 = (vc <= 0.f) && (d1 >= 0.f) && (d3 <= 0.f);
      w2 = m3 ? t3 : w2;        w3 = m3 ? 0.f : w3;  // edge AB
      bool m2 = (d3 >= 0.f) && (d4 <= d3);
      w2 = m2 ? 1.f : w2;       w3 = m2 ? 0.f : w3;  // vertex B
      bool m1 = (d1 <= 0.f) && (d2 <= 0.f);
      w2 = m1 ? 0.f : w2;       w3 = m1 ? 0.f : w3;  // vertex A
      // |p-cp|^2 - |p|^2 (monotone in dist for fixed p)
      float score = -2.f * q + w2 * (w2 * Lab - 2.f * d1)
                            + w3 * (w3 * Lac - 2.f * d2)
                            + 2.f * (w2 * w3) * Mm;
      if (score < bs[r]) { bs[r] = score; bi[r] = fcol; bw2[r] = w2; bw3[r] = w3; }
    }
  }

  // reduce across the 16 columns (stays within each 16-lane half = same row set)
  for (int off = 8; off >= 1; off >>= 1) {
#pragma unroll
    for (int r = 0; r < 8; ++r) {
      float os = __shfl_xor(bs[r], off, 32);
      int   oi = __shfl_xor(bi[r], off, 32);
      float o2 = __shfl_xor(bw2[r], off, 32);
      float o3 = __shfl_xor(bw3[r], off, 32);
      bool take = (os < bs[r]) || (os == bs[r] && oi < bi[r]);
      if (take) { bs[r] = os; bi[r] = oi; bw2[r] = o2; bw3[r] = o3; }
    }
  }
  if (col == 0) {
    int rowbase = pt_tile * 16 + (hi ? 8 : 0);
#pragma unroll
    for (int r = 0; r < 8; ++r) {
      float* o = part + ((size_t)(rowbase + r) * CHUNKS + chunk) * 4;
      o[0] = bs[r];
      ((int*)o)[1] = bi[r];
      o[2] = bw2[r];
      o[3] = bw3[r];
    }
  }
}

// ---------------------------------------------------------------------------
// Kernel 3: combine chunk partials per point; emit idx, w1, w2, w3.
// ---------------------------------------------------------------------------
__global__ void p2f_reduce_kernel(const float* __restrict__ part,
                                  float* __restrict__ out) {
  int p = blockIdx.x * blockDim.x + threadIdx.x;
  if (p >= P_N) return;
  float bs = 3.402823466e38f, b2 = 0.f, b3 = 0.f;
  int bi = 0;
  for (int c = 0; c < CHUNKS; ++c) {
    const float* e = part + ((size_t)p * CHUNKS + c) * 4;
    float s = e[0];
    int i = ((const int*)e)[1];
    if (s < bs || (s == bs && i < bi)) { bs = s; bi = i; b2 = e[2]; b3 = e[3]; }
  }
  ((int*)out)[p] = bi;                  // idx (int32 bits in float slot)
  out[P_N + p]     = 1.f - b2 - b3;     // w1
  out[2 * P_N + p] = b2;                // w2
  out[3 * P_N + p] = b3;                // w3
}

extern "C" void kernel_launch(void* const* d_in, const int* in_sizes, int n_in,
                              void* d_out, int out_size, void* d_ws, size_t ws_size,
                              hipStream_t stream) {
  const float* points   = (const float*)d_in[0];   // [P,3]
  const float* vertices = (const float*)d_in[1];   // [V,3]
  const int*   faces    = (const int*)d_in[2];     // [F,3]
  float* out = (float*)d_out;

  float* fd   = (float*)d_ws;                              // F * 16 floats = 512 KB
  float* part = (float*)((char*)d_ws + (size_t)F_N * 16 * sizeof(float)); // P*CHUNKS*4 = 512 KB

  prep_faces_kernel<<<(F_N + 255) / 256, 256, 0, stream>>>(vertices, faces, fd);
  p2f_main_kernel<<<(WAVES_TOTAL * 32) / 256, 256, 0, stream>>>(points, fd, part);
  p2f_reduce_kernel<<<(P_N + 255) / 256, 256, 0, stream>>>(part, out);
}